// SectorGCN_70549132804572
// MI455X (gfx1250) — compile-verified
//
#include <hip/hip_runtime.h>

// GCN 2-layer forward for MI455X (gfx1250, wave32).
// Memory-bound workload; GEMM h1 = x @ W1 done with V_WMMA_F32_16X16X4_F32
// (fp32 in/out, matches reference precision). Edge scatters use fp32 global
// atomics which stay L2-resident (h1/out1 are 6.4MB each, L2 = 192MB).

typedef __attribute__((ext_vector_type(2))) float v2f;
typedef __attribute__((ext_vector_type(8))) float v8f;

#define DIN 128
#define DH  16

// ---------------------------------------------------------------- degree ----
__global__ void gcn_init_deg(float* __restrict__ deg, int n) {
  int i = blockIdx.x * blockDim.x + threadIdx.x;
  if (i < n) deg[i] = 1.0f;  // self-loop contributes weight 1
}

__global__ void gcn_deg_accum(const long long* __restrict__ col,
                              const float* __restrict__ ew,
                              float* __restrict__ deg, int e) {
  int i = blockIdx.x * blockDim.x + threadIdx.x;
  if (i < e) atomicAdd(&deg[(int)col[i]], ew[i]);
}

__global__ void gcn_dinv(const float* __restrict__ deg,
                         float* __restrict__ dinv, int n) {
  int i = blockIdx.x * blockDim.x + threadIdx.x;
  if (i < n) {
    float d = deg[i];
    dinv[i] = d > 0.0f ? rsqrtf(d) : 0.0f;
  }
}

// ------------------------------------------------- GEMM h1 = x @ W1 (WMMA) --
// One wave computes one 16x16 output tile: 16 rows of x against all 16 output
// features, K=128 swept in steps of 4 via V_WMMA_F32_16X16X4_F32.
// fp32 A 16x4 layout: lanes 0-15 hold K={0,1}, lanes 16-31 hold K={2,3}.
// fp32 C/D layout: VGPR r -> M=r (lanes 0-15) / M=8+r (lanes 16-31), N=lane%16.
__global__ void gcn_gemm1_wmma(const float* __restrict__ x,
                               const float* __restrict__ W1,
                               float* __restrict__ h1, int ntiles) {
  int wave = (int)((blockIdx.x * blockDim.x + threadIdx.x) >> 5);
  int lane = threadIdx.x & 31;
  if (wave >= ntiles) return;          // wave-uniform: EXEC stays all-ones

  int row0 = wave * 16;
  int half = lane >> 4;                // 0: K pair {0,1}; 1: K pair {2,3}
  int lidx = lane & 15;

  const float* arow = x + (size_t)(row0 + lidx) * DIN;
  v8f c = {};

#pragma unroll
  for (int k = 0; k < DIN; k += 4) {
    int ka = k + 2 * half;
    v2f a, b;
    a.x = arow[ka + 0];
    a.y = arow[ka + 1];
    b.x = W1[(ka + 0) * DH + lidx];
    b.y = W1[(ka + 1) * DH + lidx];
    // (neg_a, A, neg_b, B, c_mod, C, reuse_a, reuse_b)
    c = __builtin_amdgcn_wmma_f32_16x16x4_f32(false, a, false, b,
                                              (short)0, c, false, false);
  }

#pragma unroll
  for (int r = 0; r < 8; ++r) {
    h1[(size_t)(row0 + r + 8 * half) * DH + lidx] = c[r];
  }
}

// ----------------------------------- out1 = b1 + dinv^2 * h1 (self-loops) ---
__global__ void gcn_selfloop1(const float* __restrict__ h1,
                              const float* __restrict__ dinv,
                              const float* __restrict__ b1,
                              float* __restrict__ out1, int total) {
  int i = blockIdx.x * blockDim.x + threadIdx.x;
  if (i < total) {
    int node = i >> 4;
    int f = i & (DH - 1);
    float di = dinv[node];
    out1[i] = b1[f] + di * di * h1[i];
  }
}

// ------------------------------------------- edge scatter, layer 1 ---------
// 4 lanes per edge, each handling 4 features: float4 gather + 4 fp32 atomics.
__global__ void gcn_scatter1(const long long* __restrict__ row,
                             const long long* __restrict__ col,
                             const float* __restrict__ ew,
                             const float* __restrict__ dinv,
                             const float* __restrict__ h1,
                             float* __restrict__ out1, int e) {
  int g = blockIdx.x * blockDim.x + threadIdx.x;
  int eid = g >> 2;
  if (eid >= e) return;
  int fb = (g & 3) * 4;

  __builtin_prefetch(row + eid + 4096, 0, 1);  // global_prefetch on index stream

  int r = (int)row[eid];
  int c = (int)col[eid];
  float nrm = dinv[r] * ew[eid] * dinv[c];

  float4 hv = *(const float4*)(h1 + (size_t)r * DH + fb);
  float* o = out1 + (size_t)c * DH + fb;
  atomicAdd(o + 0, nrm * hv.x);
  atomicAdd(o + 1, nrm * hv.y);
  atomicAdd(o + 2, nrm * hv.z);
  atomicAdd(o + 3, nrm * hv.w);
}

// ---------------- layer 2 per-node: h2 = relu(out1) @ W2; seed output -------
__global__ void gcn_layer2_node(const float* __restrict__ out1,
                                const float* __restrict__ W2,
                                const float* __restrict__ b2,
                                const float* __restrict__ dinv,
                                float* __restrict__ h2,
                                float* __restrict__ out, int n) {
  int i = blockIdx.x * blockDim.x + threadIdx.x;
  if (i < n) {
    const float* hrow = out1 + (size_t)i * DH;
    float acc = 0.0f;
#pragma unroll
    for (int f = 0; f < DH; ++f) {
      float v = hrow[f];
      v = v > 0.0f ? v : 0.0f;
      acc += v * W2[f];
    }
    h2[i] = acc;
    float di = dinv[i];
    out[i] = b2[0] + di * di * acc;  // bias + self-loop message
  }
}

// ------------------------------------------- edge scatter, layer 2 ---------
__global__ void gcn_scatter2(const long long* __restrict__ row,
                             const long long* __restrict__ col,
                             const float* __restrict__ ew,
                             const float* __restrict__ dinv,
                             const float* __restrict__ h2,
                             float* __restrict__ out, int e) {
  int i = blockIdx.x * blockDim.x + threadIdx.x;
  if (i < e) {
    int r = (int)row[i];
    int c = (int)col[i];
    float nrm = dinv[r] * ew[i] * dinv[c];
    atomicAdd(&out[c], nrm * h2[r]);
  }
}

// ---------------------------------------------------------------------------
extern "C" void kernel_launch(void* const* d_in, const int* in_sizes, int n_in,
                              void* d_out, int out_size, void* d_ws, size_t ws_size,
                              hipStream_t stream) {
  const float*     x   = (const float*)d_in[0];
  const long long* eix = (const long long*)d_in[1];   // int64 [2, E]
  const float*     ew  = (const float*)d_in[2];
  const float*     W1  = (const float*)d_in[3];
  const float*     b1  = (const float*)d_in[4];
  const float*     W2  = (const float*)d_in[5];
  const float*     b2  = (const float*)d_in[6];
  float*           out = (float*)d_out;

  int n = in_sizes[0] / DIN;   // 100000 (divisible by 16)
  int e = in_sizes[2];         // 3200000

  const long long* row = eix;
  const long long* col = eix + e;

  // workspace layout (floats): deg[n] | dinv[n] | h1[n*16] | out1[n*16] | h2[n]
  float* ws   = (float*)d_ws;
  float* deg  = ws;
  float* dinv = ws + (size_t)n;
  float* h1   = ws + 2 * (size_t)n;
  float* out1 = h1 + (size_t)n * DH;
  float* h2   = out1 + (size_t)n * DH;

  const int T = 256;
  gcn_init_deg  <<<(n + T - 1) / T, T, 0, stream>>>(deg, n);
  gcn_deg_accum <<<(e + T - 1) / T, T, 0, stream>>>(col, ew, deg, e);
  gcn_dinv      <<<(n + T - 1) / T, T, 0, stream>>>(deg, dinv, n);

  int ntiles = (n + 15) / 16;                       // 6250 waves
  gcn_gemm1_wmma<<<(ntiles + 7) / 8, 256, 0, stream>>>(x, W1, h1, ntiles);

  int nf = n * DH;
  gcn_selfloop1 <<<(nf + T - 1) / T, T, 0, stream>>>(h1, dinv, b1, out1, nf);

  int g1 = e * 4;                                   // 4 lanes per edge
  gcn_scatter1  <<<(g1 + T - 1) / T, T, 0, stream>>>(row, col, ew, dinv, h1, out1, e);

  gcn_layer2_node<<<(n + T - 1) / T, T, 0, stream>>>(out1, W2, b2, dinv, h2, out, n);
  gcn_scatter2  <<<(e + T - 1) / T, T, 0, stream>>>(row, col, ew, dinv, h2, out, e);
}